// HungarianMatcher_59992103191108
// MI455X (gfx1250) — compile-verified
//
#include <hip/hip_runtime.h>
#include <hip/hip_bf16.h>
#include <math.h>

typedef float v2f __attribute__((ext_vector_type(2)));
typedef float v8f __attribute__((ext_vector_type(8)));

#define B_ 128
#define Q_ 192
#define T_ 96
#define E_ 64
#define INF_ 1e30f

// ---------------------------------------------------------------------------
// Pack kernel: per-(b,q) record {p0,p1,param0..2,cx,cy,|emb|^2}
//              per-(b,t) record {feat0..4,label,|emb|^2,0}
// ---------------------------------------------------------------------------
__global__ __launch_bounds__(256) void hm_pack(
    const float* __restrict__ logits, const float* __restrict__ param,
    const float* __restrict__ center, const float* __restrict__ pemb,
    const float* __restrict__ tfeat, const int* __restrict__ tlab,
    const float* __restrict__ temb,
    float* __restrict__ qd, float* __restrict__ td)
{
  int idx = blockIdx.x * blockDim.x + threadIdx.x;
  const int NQ = B_ * Q_;
  if (idx < NQ) {
    float l0 = logits[idx * 2 + 0], l1 = logits[idx * 2 + 1];
    float m  = fmaxf(l0, l1);
    float e0 = expf(l0 - m), e1 = expf(l1 - m);
    float s  = e0 + e1;
    const float* er = pemb + (size_t)idx * E_;
    float a2 = 0.f;
    #pragma unroll
    for (int k = 0; k < E_; ++k) { float e = er[k]; a2 += e * e; }
    float* o = qd + (size_t)idx * 8;
    o[0] = e0 / s;               o[1] = e1 / s;
    o[2] = param[idx * 3 + 0];   o[3] = param[idx * 3 + 1];
    o[4] = param[idx * 3 + 2];
    o[5] = center[idx * 2 + 0];  o[6] = center[idx * 2 + 1];
    o[7] = a2;
  } else if (idx < NQ + B_ * T_) {
    int t = idx - NQ;
    const float* er = temb + (size_t)t * E_;
    float b2 = 0.f;
    #pragma unroll
    for (int k = 0; k < E_; ++k) { float e = er[k]; b2 += e * e; }
    float* o = td + (size_t)t * 8;
    o[0] = tfeat[t * 5 + 0]; o[1] = tfeat[t * 5 + 1]; o[2] = tfeat[t * 5 + 2];
    o[3] = tfeat[t * 5 + 3]; o[4] = tfeat[t * 5 + 4];
    o[5] = (float)tlab[t];
    o[6] = b2;
    o[7] = 0.f;
  }
}

// ---------------------------------------------------------------------------
// Cost kernel: 8 waves per block, one 16(q) x 16(t) tile per wave.
// Gram matrix via V_WMMA_F32_16X16X4_F32 chained over K=64; all 32 A/B
// chunk loads issued before the WMMA chain so only one loadcnt wait occurs.
// Writes CT[b][t][q] (transposed for coalesced LSA row sweeps).
// ---------------------------------------------------------------------------
__global__ __launch_bounds__(256) void hm_cost(
    const float* __restrict__ pemb, const float* __restrict__ temb,
    const float* __restrict__ qd, const float* __restrict__ td,
    float* __restrict__ CT)
{
  int wave = threadIdx.x >> 5;
  int tile = blockIdx.x * 8 + wave;
  int tt = tile % (T_ / 16);
  int qt = (tile / (T_ / 16)) % (Q_ / 16);
  int b  = tile / ((T_ / 16) * (Q_ / 16));

  int lane = threadIdx.x & 31;
  int half = lane >> 4;        // 0: K pair {0,1}, 1: K pair {2,3} per chunk
  int nloc = lane & 15;        // A row (M) / B col (N)
  int q0 = qt * 16, t0 = tt * 16;

  const float* arow = pemb + ((size_t)b * Q_ + q0 + nloc) * E_;
  const float* brow = temb + ((size_t)b * T_ + t0 + nloc) * E_;
  int k0 = 2 * half;

  // Stage all K-chunks into registers first (breaks load->wmma serialization)
  v2f a[16], bm[16];
  #pragma unroll
  for (int kk = 0; kk < 16; ++kk) {
    a[kk]  = *(const v2f*)(arow + 4 * kk + k0);
    bm[kk] = *(const v2f*)(brow + 4 * kk + k0);
  }

  v8f acc = {};
  #pragma unroll
  for (int kk = 0; kk < 16; ++kk) {
    acc = __builtin_amdgcn_wmma_f32_16x16x4_f32(
        false, a[kk], false, bm[kk], (short)0, acc, false, false);
  }

  // Epilogue: lane owns column t = t0 + nloc, rows q = q0 + 8*half + r
  int t = t0 + nloc;
  const float* tp = td + ((size_t)b * T_ + t) * 8;
  float f0 = tp[0], f1 = tp[1], f2 = tp[2], fx = tp[3], fy = tp[4];
  int   lab = (int)tp[5];
  float b2e = tp[6];
  float b2c = fx * fx + fy * fy;

  float* ob = CT + ((size_t)b * T_ + t) * Q_ + q0 + 8 * half;
  #pragma unroll
  for (int r = 0; r < 8; ++r) {
    int q = q0 + 8 * half + r;
    const float* qp = qd + ((size_t)b * Q_ + q) * 8;
    float p0 = qp[0], p1 = qp[1];
    float pr0 = qp[2], pr1 = qp[3], pr2 = qp[4];
    float cx = qp[5], cy = qp[6], a2e = qp[7];

    float cclass = -(lab ? p1 : p0);
    float cparam = fabsf(pr0 - f0) + fabsf(pr1 - f1) + fabsf(pr2 - f2);
    float a2c = cx * cx + cy * cy;
    float d2c = (a2c + b2c) - 2.0f * (cx * fx + cy * fy);
    float ccen = sqrtf(fmaxf(d2c, 0.f));
    float d2e = (a2e + b2e) - 2.0f * acc[r];
    float cemb = sqrtf(fmaxf(d2e, 0.f));

    ob[r] = 1.0f * cparam + 2.0f * cclass + 2.0f * ccen + 0.5f * cemb;
  }
}

// ---------------------------------------------------------------------------
// LSA kernel: one wave32 per batch. Jonker-Volgenant shortest augmenting
// path, lane k owns columns j = k*32+lane (6 per lane) in registers.
// ---------------------------------------------------------------------------
__global__ __launch_bounds__(32) void hm_lsa(
    const float* __restrict__ CT, float* __restrict__ outi,
    float* __restrict__ outj)
{
  __shared__ float u_s[T_];
  __shared__ float sh_s[Q_];
  __shared__ int   path_s[Q_];
  __shared__ int   r4c_s[Q_];   // row4col
  __shared__ int   c4r_s[T_];   // col4row

  int lane = threadIdx.x;
  int b = blockIdx.x;
  const float* Cb = CT + (size_t)b * T_ * Q_;

  float v[6], sh[6];
  int   pa[6];
  bool  sc[6], sr[3];

  #pragma unroll
  for (int k = 0; k < 6; ++k) { v[k] = 0.f; r4c_s[k * 32 + lane] = -1; }
  #pragma unroll
  for (int kk = 0; kk < 3; ++kk) { int t = kk * 32 + lane; u_s[t] = 0.f; c4r_s[t] = -1; }
  __syncthreads();

  for (int cur = 0; cur < T_; ++cur) {
    #pragma unroll
    for (int k = 0; k < 6; ++k) { sh[k] = INF_; pa[k] = -1; sc[k] = false; }
    sr[0] = sr[1] = sr[2] = false;
    float minVal = 0.f;
    int i = cur;
    int sink = -1;

    while (sink < 0) {
      if ((i & 31) == lane) sr[i >> 5] = true;
      float ui = u_s[i];
      const float* crow = Cb + (size_t)i * Q_;

      float bv = INF_; int bj = Q_;
      #pragma unroll
      for (int k = 0; k < 6; ++k) {
        int j = k * 32 + lane;
        float r = ((minVal + crow[j]) - ui) - v[k];
        if (!sc[k] && r < sh[k]) { sh[k] = r; pa[k] = i; }
        float mval = sc[k] ? INF_ : sh[k];
        if (mval < bv || (mval == bv && j < bj)) { bv = mval; bj = j; }
      }
      // wave-wide lexicographic (value, index) min == jnp.argmin semantics
      #pragma unroll
      for (int off = 16; off > 0; off >>= 1) {
        float ov = __shfl_xor(bv, off);
        int   oj = __shfl_xor(bj, off);
        if (ov < bv || (ov == bv && oj < bj)) { bv = ov; bj = oj; }
      }
      minVal = bv;
      int js = bj;
      if ((js & 31) == lane) sc[js >> 5] = true;
      int rj = r4c_s[js];
      if (rj < 0) sink = js; else i = rj;
    }

    // spill per-lane shortest/path for cross-lane access
    #pragma unroll
    for (int k = 0; k < 6; ++k) { int j = k * 32 + lane; sh_s[j] = sh[k]; path_s[j] = pa[k]; }
    __syncthreads();

    // dual updates (pre-augmentation state, as in reference)
    if (lane == 0) u_s[cur] += minVal;
    #pragma unroll
    for (int kk = 0; kk < 3; ++kk) {
      int t = kk * 32 + lane;
      if (sr[kk] && t != cur) {
        int c = c4r_s[t];
        c = c < 0 ? 0 : (c > Q_ - 1 ? Q_ - 1 : c);
        u_s[t] = u_s[t] + minVal - sh_s[c];
      }
    }
    #pragma unroll
    for (int k = 0; k < 6; ++k)
      if (sc[k]) v[k] = v[k] + sh[k] - minVal;
    __syncthreads();

    // augment along predecessor path
    if (lane == 0) {
      int j = sink;
      while (true) {
        int ii = path_s[j];
        r4c_s[j] = ii;
        int nj = c4r_s[ii];
        c4r_s[ii] = j;
        j = nj;
        if (ii == cur) break;
      }
    }
    __syncthreads();
  }

  // argsort by assigned query (values distinct): rank counting
  #pragma unroll
  for (int kk = 0; kk < 3; ++kk) {
    int t = kk * 32 + lane;
    int val = c4r_s[t];
    int rank = 0;
    for (int ttv = 0; ttv < T_; ++ttv) rank += (c4r_s[ttv] < val) ? 1 : 0;
    outi[b * T_ + rank] = (float)val;  // index_i: prediction index
    outj[b * T_ + rank] = (float)t;    // index_j: target index
  }
}

// ---------------------------------------------------------------------------
extern "C" void kernel_launch(void* const* d_in, const int* in_sizes, int n_in,
                              void* d_out, int out_size, void* d_ws, size_t ws_size,
                              hipStream_t stream) {
  const float* logits = (const float*)d_in[0];
  const float* param  = (const float*)d_in[1];
  const float* center = (const float*)d_in[2];
  const float* pemb   = (const float*)d_in[3];
  const float* tfeat  = (const float*)d_in[4];
  const int*   tlab   = (const int*)d_in[5];
  const float* temb   = (const float*)d_in[6];

  float* CT = (float*)d_ws;                                  // B*T*Q
  float* qd = CT + (size_t)B_ * T_ * Q_;                     // B*Q*8
  float* td = qd + (size_t)B_ * Q_ * 8;                      // B*T*8

  float* outi = (float*)d_out;
  float* outj = outi + B_ * T_;

  int npack = B_ * Q_ + B_ * T_;
  hm_pack<<<(npack + 255) / 256, 256, 0, stream>>>(
      logits, param, center, pemb, tfeat, tlab, temb, qd, td);

  int ntiles = B_ * (Q_ / 16) * (T_ / 16);   // 128 * 12 * 6 = 9216
  hm_cost<<<ntiles / 8, 256, 0, stream>>>(pemb, temb, qd, td, CT);

  hm_lsa<<<B_, 32, 0, stream>>>(CT, outi, outj);
}